// MultiSpanAllocator_58944131170660
// MI455X (gfx1250) — compile-verified
//
#include <hip/hip_runtime.h>

// ---------------------------------------------------------------------------
// MultiSpanAllocator mask: 8192x8192 f32 mask, pure store-bound (256 MiB out).
//   q<4096 (text, causal):  val = (k <= q)                (covers TT + TG=0)
//   q>=4096, k<4096:        val = 1                        (GT: is_history)
//   q>=4096, k>=4096:       val = (dist^2(grid_q,grid_k) < 6.25)
// Grid-grid dist^2 computed with V_WMMA_F32_16X16X4_F32, folding
//   dist^2 = qy*(-2ky) + qx*(-2kx) + sqq*1 + 1*sqk   into one K=4 matmul.
// ---------------------------------------------------------------------------

#define T_TEXT 4096
#define TOTAL  8192

typedef __attribute__((ext_vector_type(2))) float v2f;
typedef __attribute__((ext_vector_type(4))) float f4;
typedef __attribute__((ext_vector_type(8))) float v8f;

// ---- trivial quadrants: TT (causal tri), TG (zeros), GT (ones) -------------
// One float4 per thread, nontemporal b128 stores (streaming 192 MiB).
__global__ void __launch_bounds__(256)
mask_trivial_kernel(float* __restrict__ out) {
  unsigned tid = blockIdx.x * 256u + threadIdx.x;
  unsigned q   = tid >> 11;        // row       (8192 rows)
  unsigned c4  = tid & 2047u;      // float4 col (2048 per row)
  if (q >= T_TEXT && c4 >= 1024u) return;   // GG quadrant handled by WMMA kernel

  unsigned k0 = c4 << 2;
  f4 v;
  if (q < T_TEXT) {
    // causal: 1 where k <= q; for k >= 4096 this is 0 automatically (k > q)
    v.x = (k0      <= q) ? 1.0f : 0.0f;
    v.y = (k0 + 1u <= q) ? 1.0f : 0.0f;
    v.z = (k0 + 2u <= q) ? 1.0f : 0.0f;
    v.w = (k0 + 3u <= q) ? 1.0f : 0.0f;
  } else {
    v.x = 1.0f; v.y = 1.0f; v.z = 1.0f; v.w = 1.0f;   // grid-q over text-k
  }
  __builtin_nontemporal_store(v, (f4*)(out + (size_t)q * TOTAL + k0));
}

// ---- grid x grid quadrant via WMMA -----------------------------------------
// One wave per 16x16 tile of the 4096x4096 GG block (256x256 tiles).
// A (16x4 f32, 2 VGPRs): lanes 0-15 -> row M=lane, K=0,1 ; lanes 16-31 -> K=2,3
// B (4x16 f32, 2 VGPRs): lanes 0-15 -> col N=lane, K=0,1 ; lanes 16-31 -> K=2,3
// D (16x16 f32, 8 VGPRs): VGPR r, lanes 0-15 -> M=r, N=lane ; lanes 16-31 -> M=r+8
__global__ void __launch_bounds__(256)
mask_grid_wmma_kernel(float* __restrict__ out) {
  unsigned lane  = threadIdx.x & 31u;
  unsigned wave  = blockIdx.x * 8u + (threadIdx.x >> 5);  // 8 waves / block
  unsigned tileN = wave & 255u;
  unsigned tileM = wave >> 8;
  unsigned half  = lane >> 4;      // 0: K=0,1 slots ; 1: K=2,3 slots
  unsigned idx   = lane & 15u;

  unsigned qg = tileM * 16u + idx;          // query grid token for A row
  unsigned kg = tileN * 16u + idx;          // key grid token for B col
  float qy = (float)(qg >> 6),  qx = (float)(qg & 63u);
  float ky = (float)(kg >> 6),  kx = (float)(kg & 63u);
  float sqq = qy * qy + qx * qx;
  float sqk = ky * ky + kx * kx;

  // Branch-free operand setup so EXEC is all-1s at the WMMA.
  v2f a, b;
  a.x = half ? sqq          : qy;           // K=0 / K=2
  a.y = half ? 1.0f         : qx;           // K=1 / K=3
  b.x = half ? 1.0f         : -2.0f * ky;   // K=0 / K=2
  b.y = half ? sqk          : -2.0f * kx;   // K=1 / K=3

  v8f c = {};
  // D[m][n] = -2*qy*ky - 2*qx*kx + sqq + sqk = squared distance
  v8f d = __builtin_amdgcn_wmma_f32_16x16x4_f32(
      /*neg_a=*/false, a, /*neg_b=*/false, b,
      /*c_mod=*/(short)0, c, /*reuse_a=*/false, /*reuse_b=*/false);

  size_t base = (size_t)(T_TEXT + tileM * 16u + half * 8u) * TOTAL
              + (size_t)(T_TEXT + tileN * 16u + idx);
#pragma unroll
  for (int r = 0; r < 8; ++r) {
    float val = (d[r] < 6.25f) ? 1.0f : 0.0f;   // exact int dist^2 <= 6
    __builtin_nontemporal_store(val, out + base + (size_t)r * TOTAL);
  }
}

extern "C" void kernel_launch(void* const* d_in, const int* in_sizes, int n_in,
                              void* d_out, int out_size, void* d_ws, size_t ws_size,
                              hipStream_t stream) {
  (void)d_in; (void)in_sizes; (void)n_in; (void)out_size; (void)d_ws; (void)ws_size;
  float* out = (float*)d_out;

  // 8192 rows x 2048 float4 cols = 16,777,216 threads (GG threads early-exit).
  mask_trivial_kernel<<<65536, 256, 0, stream>>>(out);

  // 256x256 tiles, one wave each: 65536 waves = 8192 blocks x 8 waves.
  mask_grid_wmma_kernel<<<8192, 256, 0, stream>>>(out);
}